// WindowsLite2MSA_70050916597973
// MI455X (gfx1250) — compile-verified
//
#include <hip/hip_runtime.h>
#include <hip/hip_bf16.h>

typedef float v2f __attribute__((ext_vector_type(2)));
typedef float v8f __attribute__((ext_vector_type(8)));
typedef float f2  __attribute__((ext_vector_type(2)));

#define QS 20            // padded row stride (words) for s_qkv / s_pool: breaks
                         // the (24*16 % 64 == 0) bank aliasing, keeps 16B align
#define TS 260           // padded row stride for transposed attout [16][256]

// ---------------------------------------------------------------------------
// Kernel 1: fully fused window pipeline. One 256-thread block (8 wave32) per
// 4x4 window. x and y stay L2-resident (192MB L2 >= both 134MB tensors), so
// the windowed gather/scatter costs HBM bandwidth only once.
// ---------------------------------------------------------------------------
__global__ __launch_bounds__(256)
void fused_win_attn(const float* __restrict__ x,
                    const float* __restrict__ qkv_w,
                    const float* __restrict__ proj_w,
                    float* __restrict__ y)
{
    __shared__ float s_qkv[384 * QS];    // qkv conv output, padded rows (30 KB)
    __shared__ float s_pool[384 * QS];   // 5x5 avg-pooled qkv          (30 KB)
    __shared__ float s_kv[32 * 8 * 12];  // kv matrices, stride 12      (12 KB)
    __shared__ float s_xa[16 * TS];      // phase 1: x window [128][16];
                                         // later: attout^T [16][TS]    (16.6 KB)

    const int n    = blockIdx.x;          // window id = (b*64 + wy)*64 + wx
    const int wx   = n & 63;
    const int wy   = (n >> 6) & 63;
    const int b    = n >> 12;
    const int t    = threadIdx.x;
    const int lane = t & 31;
    const int wv   = t >> 5;              // wave id 0..7
    const int half = lane >> 4;           // 0: lanes 0-15, 1: lanes 16-31
    const int l16  = lane & 15;

    // roll(-2,-2) on read and roll(+2,+2) on write both map window coord
    // (i,j) -> ((wy*4+i+2) mod 256, (wx*4+j+2) mod 256)
    const int h0 = wy * 4 + 2;
    const int w0 = wx * 4 + 2;            // <= 254, w0+1 <= 255 always in-row

    // ---- warm L2 for the (reused-by-every-block) weights ----
    for (int ln = t; ln < 1024; ln += 256)          // proj_w: 128KB
        __builtin_prefetch(proj_w + ln * 32, 0, 1);
    for (int ln = t; ln < 384; ln += 256)           // qkv_w: 48KB
        __builtin_prefetch(qkv_w + ln * 32, 0, 1);

    // ---- stage x window [c=128][p=16] into LDS (xwin layout, stride 16) ----
    // w-wrap only ever splits at the 2-element boundary -> two f2 loads.
    for (int pi = t; pi < 512; pi += 256) {          // (c, row) pairs
        int c = pi >> 2, i = pi & 3;
        int h = (h0 + i) & 255;
        int rowbase = (((b * 128 + c) << 8) | h) << 8;
        f2 lo = *(const f2*)&x[rowbase + w0];
        f2 hi = *(const f2*)&x[rowbase + ((w0 + 2) & 255)];
        *(f2*)&s_xa[c * 16 + i * 4]     = lo;
        *(f2*)&s_xa[c * 16 + i * 4 + 2] = hi;
    }
    __syncthreads();

    // ---- qkv grouped conv: 4 groups x (96x32 @ 32x16) via WMMA f32 16x16x4
    // 24 tile jobs (group, m-tile) over 8 waves; dual accumulators for ILP.
    for (int job = wv; job < 24; job += 8) {
        int g0    = job / 6;
        int mt    = job % 6;
        int mbase = g0 * 96 + mt * 16;    // output channel base 0..383
        v8f acc0 = {0.f, 0.f, 0.f, 0.f, 0.f, 0.f, 0.f, 0.f};
        v8f acc1 = acc0;
        for (int k = 0; k < 32; k += 8) {
            v2f a0, a1, b0, b1;
            int arow = (mbase + l16) * 32 + 2 * half;
            a0.x = qkv_w[arow + k];
            a0.y = qkv_w[arow + k + 1];
            a1.x = qkv_w[arow + k + 4];
            a1.y = qkv_w[arow + k + 5];
            int brow = (g0 * 32 + k + 2 * half) * 16 + l16;
            b0.x = s_xa[brow];
            b0.y = s_xa[brow + 16];
            b1.x = s_xa[brow + 64];
            b1.y = s_xa[brow + 80];
            acc0 = __builtin_amdgcn_wmma_f32_16x16x4_f32(
                false, a0, false, b0, (short)0, acc0, false, false);
            acc1 = __builtin_amdgcn_wmma_f32_16x16x4_f32(
                false, a1, false, b1, (short)0, acc1, false, false);
        }
        v8f acc = acc0 + acc1;
        for (int r = 0; r < 8; ++r)
            s_qkv[(mbase + r + 8 * half) * QS + l16] = acc[r];
    }
    __syncthreads();

    // ---- 5x5 avg pool (pad 2) on 4x4 grid: per-channel, register-resident --
    for (int ch = t; ch < 384; ch += 256) {
        const float* row = &s_qkv[ch * QS];
        float4 r0 = *(const float4*)&row[0];
        float4 r1 = *(const float4*)&row[4];
        float4 r2 = *(const float4*)&row[8];
        float4 r3 = *(const float4*)&row[12];
        float4 rs[4];
        rs[0] = make_float4(r0.x + r1.x + r2.x, r0.y + r1.y + r2.y,
                            r0.z + r1.z + r2.z, r0.w + r1.w + r2.w);
        rs[1] = make_float4(rs[0].x + r3.x, rs[0].y + r3.y,
                            rs[0].z + r3.z, rs[0].w + r3.w);
        rs[2] = rs[1];
        rs[3] = make_float4(r1.x + r2.x + r3.x, r1.y + r2.y + r3.y,
                            r1.z + r2.z + r3.z, r1.w + r2.w + r3.w);
        float* out = &s_pool[ch * QS];
        for (int i = 0; i < 4; ++i) {
            float c012  = rs[i].x + rs[i].y + rs[i].z;
            float c0123 = c012 + rs[i].w;
            float c123  = rs[i].y + rs[i].z + rs[i].w;
            float4 o = make_float4(c012 / 25.0f, c0123 / 25.0f,
                                   c0123 / 25.0f, c123 / 25.0f);
            *(float4*)&out[i * 4] = o;
        }
    }
    __syncthreads();

    // ---- linear attention: groups 0..15 use qkv, 16..31 use pooled ----
    {   // step 1: kv[d][e] = sum_l relu(k[l][d]) * v[l][e]  (v[.][8] = 1)
        int g = t >> 3, d = t & 7;
        const float* base = (g < 16) ? &s_qkv[g * 24 * QS]
                                     : &s_pool[(g - 16) * 24 * QS];
        float kv[9];
        for (int e = 0; e < 9; ++e) kv[e] = 0.f;
        for (int lb = 0; lb < 4; ++lb) {
            float4 k4 = *(const float4*)&base[(8 + d) * QS + lb * 4];
            k4.x = fmaxf(k4.x, 0.f); k4.y = fmaxf(k4.y, 0.f);
            k4.z = fmaxf(k4.z, 0.f); k4.w = fmaxf(k4.w, 0.f);
            for (int e = 0; e < 8; ++e) {
                float4 v4 = *(const float4*)&base[(16 + e) * QS + lb * 4];
                kv[e] += k4.x * v4.x + k4.y * v4.y + k4.z * v4.z + k4.w * v4.w;
            }
            kv[8] += k4.x + k4.y + k4.z + k4.w;
        }
        float* kvp = &s_kv[(g * 8 + d) * 12];
        for (int e = 0; e < 9; ++e) kvp[e] = kv[e];
    }
    __syncthreads();
    {   // step 2: out[l][e] = relu(q).kv, normalized; write attout^T [l][c2]
        int g = t >> 3, idx = t & 7;
        const float* base = (g < 16) ? &s_qkv[g * 24 * QS]
                                     : &s_pool[(g - 16) * 24 * QS];
        const float4* kvb = (const float4*)&s_kv[g * 8 * 12];
        for (int pp = 0; pp < 2; ++pp) {
            int l = idx + pp * 8;
            float num[9];
            for (int e = 0; e < 9; ++e) num[e] = 0.f;
            for (int d = 0; d < 8; ++d) {
                float qd = fmaxf(base[d * QS + l], 0.f);
                float4 k0 = kvb[d * 3 + 0];
                float4 k1 = kvb[d * 3 + 1];
                float4 k2 = kvb[d * 3 + 2];
                num[0] += qd * k0.x; num[1] += qd * k0.y;
                num[2] += qd * k0.z; num[3] += qd * k0.w;
                num[4] += qd * k1.x; num[5] += qd * k1.y;
                num[6] += qd * k1.z; num[7] += qd * k1.w;
                num[8] += qd * k2.x;
            }
            float inv = 1.0f / (num[8] + 1e-15f);
            float4 o0 = make_float4(num[0] * inv, num[1] * inv,
                                    num[2] * inv, num[3] * inv);
            float4 o1 = make_float4(num[4] * inv, num[5] * inv,
                                    num[6] * inv, num[7] * inv);
            *(float4*)&s_xa[l * TS + g * 8]     = o0;
            *(float4*)&s_xa[l * TS + g * 8 + 4] = o1;
        }
    }
    __syncthreads();

    // ---- proj conv: (128x256) @ (256x16) via WMMA, wave wv owns m-tile wv,
    // dual accumulators; B read from transposed attout (stride 260 -> no
    // bank conflicts, mergeable b64 pairs).
    {
        int m0 = wv * 16;
        v8f acc0 = {0.f, 0.f, 0.f, 0.f, 0.f, 0.f, 0.f, 0.f};
        v8f acc1 = acc0;
        const float* brow = &s_xa[l16 * TS + 2 * half];
        const float* arow = &proj_w[(m0 + l16) * 256 + 2 * half];
        for (int k = 0; k < 256; k += 8) {
            v2f a0, a1, b0, b1;
            a0.x = arow[k];     a0.y = arow[k + 1];
            a1.x = arow[k + 4]; a1.y = arow[k + 5];
            b0.x = brow[k];     b0.y = brow[k + 1];
            b1.x = brow[k + 4]; b1.y = brow[k + 5];
            acc0 = __builtin_amdgcn_wmma_f32_16x16x4_f32(
                false, a0, false, b0, (short)0, acc0, false, false);
            acc1 = __builtin_amdgcn_wmma_f32_16x16x4_f32(
                false, a1, false, b1, (short)0, acc1, false, false);
        }
        v8f acc = acc0 + acc1;
        // scatter pre-BN y (roll(+2) folded into h0/w0)
        for (int r = 0; r < 8; ++r) {
            int ch = m0 + r + 8 * half;
            int p  = l16;
            int h  = (h0 + (p >> 2)) & 255;
            int w  = (w0 + (p & 3)) & 255;
            y[((((b * 128 + ch) << 8) | h) << 8) | w] = acc[r];
        }
    }
}

// ---------------------------------------------------------------------------
// Kernel 2: per-(b,ch) plane sum / sumsq (65536 elems each). No atomics, no
// workspace init required -> deterministic under graph replay.
// ---------------------------------------------------------------------------
__global__ __launch_bounds__(256)
void bn_stats(const float* __restrict__ y,
              float* __restrict__ wsum, float* __restrict__ wsq)
{
    __shared__ float rs[256];
    __shared__ float rq[256];
    const int plane = blockIdx.x;                       // b*128 + ch
    const float4* p = (const float4*)(y + ((size_t)plane << 16));
    float s = 0.f, q = 0.f;
    for (int it = 0; it < 64; ++it) {
        float4 v = p[threadIdx.x + (it << 8)];
        s += v.x + v.y + v.z + v.w;
        q += v.x * v.x + v.y * v.y + v.z * v.z + v.w * v.w;
    }
    rs[threadIdx.x] = s;
    rq[threadIdx.x] = q;
    __syncthreads();
    for (int off = 128; off > 0; off >>= 1) {
        if (threadIdx.x < (unsigned)off) {
            rs[threadIdx.x] += rs[threadIdx.x + off];
            rq[threadIdx.x] += rq[threadIdx.x + off];
        }
        __syncthreads();
    }
    if (threadIdx.x == 0) { wsum[plane] = rs[0]; wsq[plane] = rq[0]; }
}

// ---------------------------------------------------------------------------
// Kernel 3: fold partials over B, compute affine BN, normalize y in place.
// ---------------------------------------------------------------------------
__global__ __launch_bounds__(256)
void bn_apply(float* __restrict__ y,
              const float* __restrict__ wsum, const float* __restrict__ wsq,
              const float* __restrict__ gamma, const float* __restrict__ beta)
{
    const int plane = blockIdx.x >> 2;      // b*128 + ch
    const int chunk = blockIdx.x & 3;
    const int ch    = plane & 127;
    float s = 0.f, q = 0.f;
    for (int bb = 0; bb < 4; ++bb) {
        s += wsum[bb * 128 + ch];
        q += wsq[bb * 128 + ch];
    }
    const float invN = 1.0f / 262144.0f;    // B*H*W
    float mean  = s * invN;
    float var   = q * invN - mean * mean;
    float scale = gamma[ch] * rsqrtf(var + 1e-5f);
    float bias  = beta[ch] - mean * scale;

    float4* p = (float4*)(y + ((size_t)plane << 16) + ((size_t)chunk << 14));
    for (int it = 0; it < 16; ++it) {
        float4 v = p[threadIdx.x + (it << 8)];
        v.x = v.x * scale + bias;
        v.y = v.y * scale + bias;
        v.z = v.z * scale + bias;
        v.w = v.w * scale + bias;
        p[threadIdx.x + (it << 8)] = v;
    }
}

// ---------------------------------------------------------------------------
extern "C" void kernel_launch(void* const* d_in, const int* in_sizes, int n_in,
                              void* d_out, int out_size, void* d_ws, size_t ws_size,
                              hipStream_t stream)
{
    const float* x      = (const float*)d_in[0];   // (4,128,256,256)
    const float* qkv_w  = (const float*)d_in[1];   // (384,32,1,1)
    const float* proj_w = (const float*)d_in[2];   // (128,256,1,1)
    const float* gamma  = (const float*)d_in[3];   // (128,)
    const float* beta   = (const float*)d_in[4];   // (128,)
    float*       y      = (float*)d_out;           // (4,128,256,256)

    float* wsum = (float*)d_ws;        // 512 floats
    float* wsq  = wsum + 512;          // 512 floats

    fused_win_attn<<<dim3(16384), dim3(256), 0, stream>>>(x, qkv_w, proj_w, y);
    bn_stats<<<dim3(512), dim3(256), 0, stream>>>(y, wsum, wsq);
    bn_apply<<<dim3(2048), dim3(256), 0, stream>>>(y, wsum, wsq, gamma, beta);
}